// GramsEmbedding_67061619360336
// MI455X (gfx1250) — compile-verified
//
#include <hip/hip_runtime.h>

// GramsEmbedding (K=2): out[tok,:] = W[idx0[tok],:] + (idx0!=idx1) * W[idx1[tok],:]
// tok = b*SEQ + s, EMBED_DIM = 128, weight is [32000, 128] f32.
//
// Mapping: one token per wave32; lane l handles 4 consecutive floats (native
// 4xf32 vector) -> each row gather is a single coalesced global_load_b128 per
// wave. Branchless union-of-one-hots via multiply mask so both gathers issue
// back-to-back and overlap latency. Non-temporal b128 store for the streamed
// output (never re-read on device).

typedef float v4f __attribute__((ext_vector_type(4)));

#define EMBED_V4 32          // 128 floats / 4
#define WAVES_PER_BLOCK 8
#define BLOCK_THREADS (WAVES_PER_BLOCK * 32)

__global__ __launch_bounds__(BLOCK_THREADS)
void grams_embed_kernel(const int* __restrict__ idx,      // [2, tokens] flat
                        const float* __restrict__ weight, // [32000, 128]
                        float* __restrict__ out,          // [tokens, 128]
                        int tokens) {
    const int lane = threadIdx.x & 31;
    const int wave = threadIdx.x >> 5;
    const int tok  = blockIdx.x * WAVES_PER_BLOCK + wave;
    if (tok >= tokens) return;

    // Two gram indices for this token (wave-uniform values; one cacheline
    // request per wave since all 32 lanes use the same address).
    const int i0 = idx[tok];
    const int i1 = idx[tokens + tok];

    const v4f* __restrict__ w4 = (const v4f*)weight;

    // Issue both row gathers immediately so the two global_load_b128 latencies
    // overlap (weight table lives in the 192MB L2 after first touch).
    const v4f a = w4[(long)i0 * EMBED_V4 + lane];
    const v4f b = w4[(long)i1 * EMBED_V4 + lane];

    // Union semantics of the n-hot: duplicate gram must not be double counted.
    const float m = (i0 != i1) ? 1.0f : 0.0f;

    v4f r;
    r.x = fmaf(m, b.x, a.x);
    r.y = fmaf(m, b.y, a.y);
    r.z = fmaf(m, b.z, a.z);
    r.w = fmaf(m, b.w, a.w);

    // Output is write-once / never re-read on device: non-temporal (TH=NT)
    // store keeps it out of the caches.
    __builtin_nontemporal_store(r, (v4f*)out + (long)tok * EMBED_V4 + lane);
}

extern "C" void kernel_launch(void* const* d_in, const int* in_sizes, int n_in,
                              void* d_out, int out_size, void* d_ws, size_t ws_size,
                              hipStream_t stream) {
    const int*   idx    = (const int*)d_in[0];    // [K=2, B, S] int32
    const float* weight = (const float*)d_in[1];  // [32000, 128] f32
    float*       out    = (float*)d_out;          // [B, S, 128] f32

    const int tokens = in_sizes[0] / 2;           // B*S (K_GRAMS == 2)
    const int blocks = (tokens + WAVES_PER_BLOCK - 1) / WAVES_PER_BLOCK;

    grams_embed_kernel<<<blocks, BLOCK_THREADS, 0, stream>>>(idx, weight, out, tokens);
}